// mAtt_bci_38199439131097
// MI455X (gfx1250) — compile-verified
//
#include <hip/hip_runtime.h>
#include <hip/hip_bf16.h>

typedef float v2f __attribute__((ext_vector_type(2)));
typedef float v8f __attribute__((ext_vector_type(8)));

#define BS   2048
#define C1   22
#define T1   438
#define C2   20
#define T2   439
#define KW   12
#define NOUT 18   // N_OUT
#define CT   64   // conv2 time-tile per block
#define LOG_RECT_EPS (-9.210340371976182f)  // ln(1e-4)

// ---------------------------------------------------------------------------
// CDNA5 f32 WMMA 16x16x4 helpers (wave32; A 16x4, B 4x16, C/D 16x16 f32)
// A-frag: lanes 0-15 hold (M=lane, K=k0..k0+1), lanes 16-31 hold (M=lane-16, K=k0+2..k0+3)
// B-frag: lanes 0-15 hold (N=lane, K=k0/k0+1 in VGPR0/1), lanes 16-31 K=k0+2/k0+3
// D-frag: VGPR r -> row = r + 8*(lane>=16), col = lane&15
// ---------------------------------------------------------------------------
__device__ __forceinline__ v8f wmma_f32_16x16x4(v2f a, v2f b, v8f c) {
  return __builtin_amdgcn_wmma_f32_16x16x4_f32(false, a, false, b, (short)0, c, false, false);
}

__device__ __forceinline__ v2f ldfragA(const float* S, int ld, int m0, int k0, int lane) {
  int m = m0 + (lane & 15);
  int k = k0 + ((lane >> 4) << 1);
  v2f r;
  r.x = S[m * ld + k];
  r.y = S[m * ld + k + 1];
  return r;
}
__device__ __forceinline__ v2f ldfragB(const float* S, int ld, int k0, int n0, int lane) {
  int n = n0 + (lane & 15);
  int k = k0 + ((lane >> 4) << 1);
  v2f r;
  r.x = S[k * ld + n];
  r.y = S[(k + 1) * ld + n];
  return r;
}
__device__ __forceinline__ void stfragD(float* S, int ld, int m0, int n0, v8f acc, int lane,
                                        int rmax, int cmax, float scale) {
  int col = n0 + (lane & 15);
  int rbase = m0 + ((lane >> 4) << 3);
#pragma unroll
  for (int r = 0; r < 8; ++r) {
    int row = rbase + r;
    if (row < rmax && col < cmax) S[row * ld + col] = acc[r] * scale;
  }
}

// intra-wave LDS producer->consumer fence (CDNA5 split counters)
__device__ __forceinline__ void wave_lds_fence() {
  __builtin_amdgcn_wave_barrier();
  asm volatile("s_wait_dscnt 0" ::: "memory");
  __builtin_amdgcn_wave_barrier();
}

// ---------------------------------------------------------------------------
// Wave32 cyclic Jacobi eigensolver for symmetric 18x18.
// Lane j (j<18) owns column j: a[i] = A[i][j], u[i] = U[i][j].
// ---------------------------------------------------------------------------
__device__ void jacobi18(float a[NOUT], float u[NOUT], int lane) {
#pragma unroll
  for (int i = 0; i < NOUT; ++i) u[i] = (i == lane) ? 1.0f : 0.0f;
  for (int sweep = 0; sweep < 8; ++sweep) {
    for (int p = 0; p < NOUT - 1; ++p) {
      for (int q = p + 1; q < NOUT; ++q) {
        float apj = 0.f, aqj = 0.f;
#pragma unroll
        for (int i = 0; i < NOUT; ++i) {
          if (i == p) apj = a[i];
          if (i == q) aqj = a[i];
        }
        float app = __shfl(apj, p, 32);
        float apq = __shfl(apj, q, 32);
        float aqq = __shfl(aqj, q, 32);
        float c = 1.f, s = 0.f;
        if (fabsf(apq) > 1e-12f) {  // uniform across wave (broadcast values)
          float tau = (aqq - app) / (2.f * apq);
          float t = copysignf(1.f, tau) / (fabsf(tau) + sqrtf(1.f + tau * tau));
          c = rsqrtf(1.f + t * t);
          s = t * c;
        }
        // row rotation: A <- G^T A  (rows p,q; per-lane on its column)
        float np = c * apj - s * aqj;
        float nq = s * apj + c * aqj;
#pragma unroll
        for (int i = 0; i < NOUT; ++i) {
          if (i == p) a[i] = np;
          if (i == q) a[i] = nq;
        }
        // column rotation: A <- A G  (cols p,q; lanes p,q) + U <- U G
#pragma unroll
        for (int i = 0; i < NOUT; ++i) {
          float vp = __shfl(a[i], p, 32);
          float vq = __shfl(a[i], q, 32);
          float t0 = c * vp - s * vq;
          float t1 = s * vp + c * vq;
          if (lane == p) a[i] = t0;
          if (lane == q) a[i] = t1;
          float up = __shfl(u[i], p, 32);
          float uq = __shfl(u[i], q, 32);
          float t2 = c * up - s * uq;
          float t3 = s * up + c * uq;
          if (lane == p) u[i] = t2;
          if (lane == q) u[i] = t3;
        }
      }
    }
  }
}

// ---------------------------------------------------------------------------
// Stage 1: conv1 (22ch <- 22 taps over H) + BN1 stat accumulation
// ---------------------------------------------------------------------------
__global__ __launch_bounds__(256) void init_stats_kernel(float* stats1, float* stats2) {
  int t = threadIdx.x;
  if (t < 2 * C1) stats1[t] = 0.f;
  if (t < 2 * C2) stats2[t] = 0.f;
}

__global__ __launch_bounds__(256) void conv1_kernel(const float* __restrict__ x,
                                                    const float* __restrict__ W1,
                                                    const float* __restrict__ b1,
                                                    float* __restrict__ h1,
                                                    float* __restrict__ stats1) {
  __shared__ float w[C1 * C1];
  __shared__ float ssum[C1], ssq[C1];
  int tid = threadIdx.x;
  if (tid < C1) { ssum[tid] = 0.f; ssq[tid] = 0.f; }
  for (int i = tid; i < C1 * C1; i += 256) w[i] = W1[i];
  __syncthreads();
  long idx = (long)blockIdx.x * 256 + tid;
  if (idx < (long)BS * T1) {
    int b = (int)(idx / T1), t = (int)(idx % T1);
    const float* xp = x + (long)b * C1 * T1 + t;
    // speculative prefetch of next block's window (global_prefetch_b8)
    long nidx = idx + 256;
    if (nidx < (long)BS * T1) {
      int nb = (int)(nidx / T1), nt = (int)(nidx % T1);
      __builtin_prefetch(x + (long)nb * C1 * T1 + nt, 0, 1);
    }
    float xv[C1];
#pragma unroll
    for (int k = 0; k < C1; ++k) xv[k] = xp[k * T1];
    for (int c = 0; c < C1; ++c) {
      float acc = b1[c];
#pragma unroll
      for (int k = 0; k < C1; ++k) acc += xv[k] * w[c * C1 + k];
      h1[((long)b * C1 + c) * T1 + t] = acc;
      atomicAdd(&ssum[c], acc);
      atomicAdd(&ssq[c], acc * acc);
    }
  }
  __syncthreads();
  if (tid < C1) {
    atomicAdd(&stats1[tid], ssum[tid]);
    atomicAdd(&stats1[C1 + tid], ssq[tid]);
  }
}

__global__ void finalize1_kernel(const float* stats1, const float* g1, const float* be1,
                                 float* sc1) {
  int c = threadIdx.x;
  if (c < C1) {
    float n = (float)BS * (float)T1;
    float mean = stats1[c] / n;
    float var = stats1[C1 + c] / n - mean * mean;
    float s = g1[c] * rsqrtf(var + 1e-5f);
    sc1[c] = s;
    sc1[C1 + c] = be1[c] - mean * s;
  }
}

// ---------------------------------------------------------------------------
// Stage 2: BN1-affine + conv2 as im2col GEMM on f32 WMMA.
// D(20x16) = W2(20x264) * col(264x16); one batch x 64-wide time tile per
// block; 8 waves = 2 M-tiles x 4 N-tiles; 66 wmma per wave (K=264).
// ---------------------------------------------------------------------------
__global__ __launch_bounds__(256) void conv2_wmma_kernel(const float* __restrict__ h1,
                                                         const float* __restrict__ sc1,
                                                         const float* __restrict__ W2,
                                                         const float* __restrict__ b2,
                                                         float* __restrict__ h2,
                                                         float* __restrict__ stats2) {
  __shared__ float Wbuf[32][C1 * KW];  // 32x264, rows >=20 zero
  __shared__ float hbuf[C1][80];       // BN1-affined window, zero-padded halo
  __shared__ float ssum[C2], ssq[C2];
  const int tid = threadIdx.x;
  const int lane = tid & 31;
  const int wave = tid >> 5;
  const int b = blockIdx.x;
  const int t0 = blockIdx.y * CT;

  if (tid < C2) { ssum[tid] = 0.f; ssq[tid] = 0.f; }
  for (int i = tid; i < 32 * C1 * KW; i += 256)
    (&Wbuf[0][0])[i] = (i < C2 * C1 * KW) ? W2[i] : 0.f;
  for (int i = tid; i < C1 * 80; i += 256) {
    int ch = i / 80, d = i % 80;
    int tau = t0 - 6 + d;
    float v = 0.f;
    if (d < 76 && tau >= 0 && tau < T1)
      v = h1[((long)b * C1 + ch) * T1 + tau] * sc1[ch] + sc1[C1 + ch];
    hbuf[ch][d] = v;
  }
  __syncthreads();

  const int m0 = (wave & 1) << 4;            // M tile (channels)
  const int n0 = (wave >> 1) << 4;           // N tile (time)
  const int nn = n0 + (lane & 15);
  const int ksel = (lane >> 4) << 1;
  const float* wrow = &Wbuf[m0 + (lane & 15)][0];
  v8f acc = {};
  for (int ch = 0; ch < C1; ++ch) {
    const float* hrow = &hbuf[ch][0];
#pragma unroll
    for (int kk0 = 0; kk0 < KW; kk0 += 4) {  // K aligned to the 12-tap kernel
      v2f av, bv;
      av.x = wrow[ch * KW + kk0 + ksel];
      av.y = wrow[ch * KW + kk0 + ksel + 1];
      bv.x = hrow[nn + kk0 + ksel];
      bv.y = hrow[nn + kk0 + ksel + 1];
      acc = wmma_f32_16x16x4(av, bv, acc);
    }
  }

  // bias + store + BN2 stats straight from the D-fragment
  const int col = n0 + (lane & 15);
  const int t = t0 + col;
  const int rbase = m0 + ((lane >> 4) << 3);
#pragma unroll
  for (int r = 0; r < 8; ++r) {
    int c = rbase + r;
    if (c < C2 && t < T2) {
      float v = acc[r] + b2[c];
      h2[((long)b * C2 + c) * T2 + t] = v;
      atomicAdd(&ssum[c], v);
      atomicAdd(&ssq[c], v * v);
    }
  }
  __syncthreads();
  if (tid < C2) {
    atomicAdd(&stats2[tid], ssum[tid]);
    atomicAdd(&stats2[C2 + tid], ssq[tid]);
  }
}

__global__ void finalize2_kernel(const float* stats2, const float* g2, const float* be2,
                                 float* sc2) {
  int c = threadIdx.x;
  if (c < C2) {
    float n = (float)BS * (float)T2;
    float mean = stats2[c] / n;
    float var = stats2[C2 + c] / n - mean * mean;
    float s = g2[c] * rsqrtf(var + 1e-5f);
    sc2[c] = s;
    sc2[C2 + c] = be2[c] - mean * s;
  }
}

// ---------------------------------------------------------------------------
// Stage 3: one workgroup (8 wave32s) per batch element: covariance SPD,
// congruence, matrix-log attention, fused exp/rect/log, triu GEMM.
// ---------------------------------------------------------------------------
struct SpdSmem {
  float scratch[3 * 32 * 148];       // phase overlay: ctr | Tb | Ub+Vb
  float Wb[3][C2][32];               // Wq/Wk/Wv, padded cols (zeros >=18)
  float Xb[3][32][32];               // covariance SPD, padded
  float QKV[9][NOUT][NOUT + 1];      // congruence outputs
  float logb[9][NOUT][NOUT + 1];     // logQ(0-2) logK(3-5) logV(6-8)
  float mixedb[3][NOUT][NOUT + 1];
  float logO[3][NOUT][NOUT + 1];
  float Em[3][3];
  float Pm[3][3];
  float means[3][C2];
  float invTr[3];
};

__global__ __launch_bounds__(256) void spd_kernel(const float* __restrict__ h2,
                                                  const float* __restrict__ sc2,
                                                  const float* __restrict__ Wq,
                                                  const float* __restrict__ Wk,
                                                  const float* __restrict__ Wv,
                                                  const float* __restrict__ Wl,
                                                  const float* __restrict__ bl,
                                                  float* __restrict__ out) {
  __shared__ SpdSmem sm;
  const int tid = threadIdx.x;
  const int lane = tid & 31;
  const int wave = tid >> 5;
  const int b = blockIdx.x;

  {  // zero all shared memory (pads must be zero for WMMA tiles)
    float* p = (float*)&sm;
    const int n = (int)(sizeof(SpdSmem) / 4);
    for (int i = tid; i < n; i += 256) p[i] = 0.f;
  }
  __syncthreads();

  const int Lp[3] = {147, 146, 146};
  const int off[3] = {0, 147, 293};
  float (*ctr)[32][148] = (float (*)[32][148])sm.scratch;

  // load Wq/Wk/Wv padded
  for (int i = tid; i < 3 * C2 * NOUT; i += 256) {
    int w = i / (C2 * NOUT);
    int r = (i / NOUT) % C2;
    int c = i % NOUT;
    const float* W = (w == 0) ? Wq : (w == 1) ? Wk : Wv;
    sm.Wb[w][r][c] = W[r * NOUT + c];
  }
  // load sig with BN2 affine into patch-split buffer
  for (int i = tid; i < C2 * T2; i += 256) {
    int c = i / T2, t = i % T2;
    int p = (t < 147) ? 0 : (t < 293) ? 1 : 2;
    float v = h2[((long)b * C2 + c) * T2 + t] * sc2[c] + sc2[C2 + c];
    ctr[p][c][t - off[p]] = v;
  }
  __syncthreads();

  // per-(patch,channel) mean, then center
  if (tid < 3 * C2) {
    int p = tid / C2, c = tid % C2;
    float s = 0.f;
    for (int i = 0; i < Lp[p]; ++i) s += ctr[p][c][i];
    sm.means[p][c] = s / (float)Lp[p];
  }
  __syncthreads();
  for (int i = tid; i < 3 * C2 * 148; i += 256) {
    int p = i / (C2 * 148);
    int c = (i / 148) % C2;
    int k = i % 148;
    if (k < Lp[p]) ctr[p][c][k] -= sm.means[p][c];
  }
  __syncthreads();

  // covariance: cov = Xc*Xc^T/(L-1) via f32 WMMA, 12 (patch,16x16-tile) tasks
  const float invL1[3] = {1.f / 146.f, 1.f / 145.f, 1.f / 145.f};
  for (int task = wave; task < 12; task += 8) {
    int p = task >> 2;
    int m0 = ((task >> 1) & 1) << 4;
    int n0 = (task & 1) << 4;
    v8f acc = {};
    for (int k0 = 0; k0 < 148; k0 += 4) {
      v2f av = ldfragA(&ctr[p][0][0], 148, m0, k0, lane);
      v2f bv = ldfragA(&ctr[p][0][0], 148, n0, k0, lane);  // B = Xc^T: same access
      acc = wmma_f32_16x16x4(av, bv, acc);
    }
    stfragD(&sm.Xb[p][0][0], 32, m0, n0, acc, lane, 32, 32, invL1[p]);
  }
  __syncthreads();

  // trace normalize + eps*I
  if (tid < 3) {
    float tr = 0.f;
    for (int c = 0; c < C2; ++c) tr += sm.Xb[tid][c][c];
    sm.invTr[tid] = 1.f / tr;
  }
  __syncthreads();
  for (int i = tid; i < 3 * 32 * 32; i += 256) {
    int p = i >> 10, r = (i >> 5) & 31, c = i & 31;
    float v = sm.Xb[p][r][c] * sm.invTr[p];
    if (r == c && r < C2) v += 1e-5f;
    sm.Xb[p][r][c] = v;
  }
  __syncthreads();

  // congruence Q/K/V = W^T X W via two WMMA GEMMs (9 tasks, waves 0..3)
  float (*Tb)[32][32] = (float (*)[32][32])sm.scratch;
  if (wave < 4) {
    for (int task = wave; task < 9; task += 4) {
      int kind = task / 3, m = task % 3;
      for (int t2 = 0; t2 < 4; ++t2) {  // T = X[m] * W[kind]
        int m0 = (t2 >> 1) << 4, n0 = (t2 & 1) << 4;
        v8f acc = {};
        for (int k0 = 0; k0 < 20; k0 += 4) {
          v2f av = ldfragA(&sm.Xb[m][0][0], 32, m0, k0, lane);
          v2f bv = ldfragB(&sm.Wb[kind][0][0], 32, k0, n0, lane);
          acc = wmma_f32_16x16x4(av, bv, acc);
        }
        stfragD(&Tb[wave][0][0], 32, m0, n0, acc, lane, 32, 32, 1.f);
      }
      wave_lds_fence();
      for (int t2 = 0; t2 < 4; ++t2) {  // Q = W^T * T
        int m0 = (t2 >> 1) << 4, n0 = (t2 & 1) << 4;
        v8f acc = {};
        for (int k0 = 0; k0 < 20; k0 += 4) {
          v2f av = ldfragB(&sm.Wb[kind][0][0], 32, k0, m0, lane);  // A = W^T
          v2f bv = ldfragB(&Tb[wave][0][0], 32, k0, n0, lane);
          acc = wmma_f32_16x16x4(av, bv, acc);
        }
        stfragD(&sm.QKV[task][0][0], NOUT + 1, m0, n0, acc, lane, NOUT, NOUT, 1.f);
      }
      wave_lds_fence();
    }
  }
  __syncthreads();

  // repurpose scratch as per-wave U / f(Lam)*U^T WMMA operands
  for (int i = tid; i < 8 * 32 * 20 + 8 * 20 * 32; i += 256) sm.scratch[i] = 0.f;
  __syncthreads();
  float (*Ub)[32][20] = (float (*)[32][20])sm.scratch;
  float (*Vb)[20][32] = (float (*)[20][32])(sm.scratch + 8 * 32 * 20);

  // eigh + matrix-log reconstruction for the 9 Q/K/V matrices
  for (int task = wave; task < 9; task += 8) {
    float a[NOUT], u[NOUT];
#pragma unroll
    for (int i = 0; i < NOUT; ++i) a[i] = (lane < NOUT) ? sm.QKV[task][i][lane] : 0.f;
    jacobi18(a, u, lane);
    float lam = 0.f;
#pragma unroll
    for (int i = 0; i < NOUT; ++i)
      if (i == lane) lam = a[i];
    float fl = logf(fmaxf(lam, 1e-30f));
    if (lane < NOUT) {
#pragma unroll
      for (int i = 0; i < NOUT; ++i) {
        Ub[wave][i][lane] = u[i];
        Vb[wave][lane][i] = fl * u[i];
      }
    }
    wave_lds_fence();
    for (int t2 = 0; t2 < 4; ++t2) {  // R = U * (f(Lam) U^T)
      int m0 = (t2 >> 1) << 4, n0 = (t2 & 1) << 4;
      v8f acc = {};
      for (int k0 = 0; k0 < 20; k0 += 4) {
        v2f av = ldfragA(&Ub[wave][0][0], 20, m0, k0, lane);
        v2f bv = ldfragB(&Vb[wave][0][0], 32, k0, n0, lane);
        acc = wmma_f32_16x16x4(av, bv, acc);
      }
      stfragD(&sm.logb[task][0][0], NOUT + 1, m0, n0, acc, lane, NOUT, NOUT, 1.f);
    }
    wave_lds_fence();
  }
  __syncthreads();

  // attention: energy, scores, softmax
  if (tid < 9) {
    int j = tid / 3, i2 = tid % 3;
    float e = 0.f;
    for (int r = 0; r < NOUT; ++r)
      for (int c = 0; c < NOUT; ++c) {
        float d = sm.logb[j][r][c] - sm.logb[3 + i2][r][c];
        e += d * d;
      }
    sm.Em[j][i2] = 1.f / (1.f + log1pf(e));
  }
  __syncthreads();
  if (tid < 3) {
    float m0 = fmaxf(sm.Em[tid][0], fmaxf(sm.Em[tid][1], sm.Em[tid][2]));
    float e0 = expf(sm.Em[tid][0] - m0);
    float e1 = expf(sm.Em[tid][1] - m0);
    float e2 = expf(sm.Em[tid][2] - m0);
    float s = e0 + e1 + e2;
    sm.Pm[tid][0] = e0 / s;
    sm.Pm[tid][1] = e1 / s;
    sm.Pm[tid][2] = e2 / s;
  }
  __syncthreads();
  for (int i = tid; i < 3 * NOUT * NOUT; i += 256) {
    int j = i / (NOUT * NOUT);
    int r = (i / NOUT) % NOUT;
    int c = i % NOUT;
    sm.mixedb[j][r][c] = sm.Pm[j][0] * sm.logb[6][r][c] + sm.Pm[j][1] * sm.logb[7][r][c] +
                         sm.Pm[j][2] * sm.logb[8][r][c];
  }
  __syncthreads();

  // fused expm -> rect -> logm: single eigh, lambda -> max(lambda, ln(eps))
  for (int m = wave; m < 3; m += 8) {
    float a[NOUT], u[NOUT];
#pragma unroll
    for (int i = 0; i < NOUT; ++i) a[i] = (lane < NOUT) ? sm.mixedb[m][i][lane] : 0.f;
    jacobi18(a, u, lane);
    float lam = 0.f;
#pragma unroll
    for (int i = 0; i < NOUT; ++i)
      if (i == lane) lam = a[i];
    float fl = fmaxf(lam, LOG_RECT_EPS);
    if (lane < NOUT) {
#pragma unroll
      for (int i = 0; i < NOUT; ++i) {
        Ub[wave][i][lane] = u[i];
        Vb[wave][lane][i] = fl * u[i];
      }
    }
    wave_lds_fence();
    for (int t2 = 0; t2 < 4; ++t2) {
      int m0 = (t2 >> 1) << 4, n0 = (t2 & 1) << 4;
      v8f acc = {};
      for (int k0 = 0; k0 < 20; k0 += 4) {
        v2f av = ldfragA(&Ub[wave][0][0], 20, m0, k0, lane);
        v2f bv = ldfragB(&Vb[wave][0][0], 32, k0, n0, lane);
        acc = wmma_f32_16x16x4(av, bv, acc);
      }
      stfragD(&sm.logO[m][0][0], NOUT + 1, m0, n0, acc, lane, NOUT, NOUT, 1.f);
    }
    wave_lds_fence();
  }
  __syncthreads();

  // triu-vectorize (row-major upper triangle) + 513x4 output GEMM
  if (tid < 4) {
    float acc = bl[tid];
    int v = 0;
    for (int m = 0; m < 3; ++m)
      for (int i = 0; i < NOUT; ++i)
        for (int j = i; j < NOUT; ++j) {
          acc += sm.logO[m][i][j] * Wl[v * 4 + tid];
          ++v;
        }
    out[(long)b * 4 + tid] = acc;
  }
}

// ---------------------------------------------------------------------------
extern "C" void kernel_launch(void* const* d_in, const int* in_sizes, int n_in,
                              void* d_out, int out_size, void* d_ws, size_t ws_size,
                              hipStream_t stream) {
  const float* x   = (const float*)d_in[0];
  const float* W1  = (const float*)d_in[1];
  const float* b1  = (const float*)d_in[2];
  const float* g1  = (const float*)d_in[3];
  const float* be1 = (const float*)d_in[4];
  const float* W2  = (const float*)d_in[5];
  const float* b2  = (const float*)d_in[6];
  const float* g2  = (const float*)d_in[7];
  const float* be2 = (const float*)d_in[8];
  const float* Wq  = (const float*)d_in[9];
  const float* Wk  = (const float*)d_in[10];
  const float* Wv  = (const float*)d_in[11];
  const float* Wl  = (const float*)d_in[12];
  const float* bl  = (const float*)d_in[13];
  float* out = (float*)d_out;

  float* ws = (float*)d_ws;
  size_t n_h1 = (size_t)BS * C1 * T1;
  size_t n_h2 = (size_t)BS * C2 * T2;
  float* h1 = ws;
  float* h2 = h1 + n_h1;
  float* stats1 = h2 + n_h2;
  float* sc1 = stats1 + 2 * C1;
  float* stats2 = sc1 + 2 * C1;
  float* sc2 = stats2 + 2 * C2;

  init_stats_kernel<<<1, 64, 0, stream>>>(stats1, stats2);
  int nb1 = (int)(((long)BS * T1 + 255) / 256);
  conv1_kernel<<<nb1, 256, 0, stream>>>(x, W1, b1, h1, stats1);
  finalize1_kernel<<<1, 32, 0, stream>>>(stats1, g1, be1, sc1);
  dim3 g2grid(BS, (T2 + CT - 1) / CT);
  conv2_wmma_kernel<<<g2grid, 256, 0, stream>>>(h1, sc1, W2, b2, h2, stats2);
  finalize2_kernel<<<1, 32, 0, stream>>>(stats2, g2, be2, sc2);
  spd_kernel<<<BS, 256, 0, stream>>>(h2, sc2, Wq, Wk, Wv, Wl, bl, out);
}